// GPNLLLoss_83923660963880
// MI455X (gfx1250) — compile-verified
//
#include <hip/hip_runtime.h>
#include <hip/hip_bf16.h>
#include <math.h>

// ---------------------------------------------------------------------------
// GP NLL loss: 128 x (1024x1024 Cholesky + forward solve), fp32.
// Trailing SYRK on V_WMMA_F32_16X16X4_F32, 2x2 macro-tiles per wave:
// 16 unguarded WMMAs per macro-tile, 4 independent accumulator chains,
// 2x fragment reuse from LDS. Guard-free hot loop via padded leading
// dimension (LDK=1040): phantom sub-tiles fall into padding / the never-read
// upper triangle, so EXEC stays all-ones with zero branches around WMMA.
// ---------------------------------------------------------------------------

typedef float v2f __attribute__((ext_vector_type(2)));
typedef float v8f __attribute__((ext_vector_type(8)));

#define NPT 1024              // N = (128/4)*(128/4)
#define LDK 1040              // padded leading dimension (+1 block col)
#define MATSZ ((size_t)LDK * LDK)   // padded per-matrix slot (rows 0..1039)
#define NBLK 64               // 1024 / 16
#define NROWS 128             // B*C
#define LOG2PI 1.8378770664093453f

// ---------------- prep: 4x4 avg pool -> residual r and sigma^2 -------------
__global__ void prep_kernel(const float* __restrict__ target,
                            const float* __restrict__ mu,
                            const float* __restrict__ sigma,
                            float* __restrict__ r_all,
                            float* __restrict__ s2_all) {
    const int bc = blockIdx.x;                 // 0..127
    const size_t ibase = (size_t)bc * 128 * 128;
    for (int n = threadIdx.x; n < NPT; n += blockDim.x) {
        const int i = n >> 5, j = n & 31;
        float ts = 0.f, ms = 0.f, ss = 0.f;
        for (int di = 0; di < 4; ++di)
            for (int dj = 0; dj < 4; ++dj) {
                const size_t idx = ibase + (size_t)(4 * i + di) * 128 + (4 * j + dj);
                ts += target[idx]; ms += mu[idx]; ss += sigma[idx];
            }
        const float rr = (ts - ms) * (1.f / 16.f);
        const float s  = ss * (1.f / 16.f);
        r_all[bc * NPT + n]  = rr;
        s2_all[bc * NPT + n] = s * s;
    }
}

// ---------------- K_base = RBF + L L^T + jitter*I --------------------------
__global__ void kbase_kernel(const float* __restrict__ L, float* __restrict__ Kb) {
    const int idx = blockIdx.x * 256 + threadIdx.x;   // 1024*1024 elements
    const int n1 = idx >> 10, n2 = idx & 1023;
    // coords[n] = ((n/32)/31, (n%32)/31)
    const float dx = (float)((n1 >> 5) - (n2 >> 5)) * (1.0f / 31.0f);
    const float dy = (float)((n1 & 31) - (n2 & 31)) * (1.0f / 31.0f);
    const float d2 = dx * dx + dy * dy;
    const float ksp = __expf(-0.5f * d2 * (1.0f / (0.1f * 0.1f)));  // OUTPUTSCALE=1
    float dot = 0.f;
#pragma unroll
    for (int t = 0; t < 16; ++t) dot += L[n1 * 16 + t] * L[n2 * 16 + t];
    Kb[(size_t)n1 * NPT + n2] = ksp + dot + ((n1 == n2) ? 1e-4f : 0.f);
}

// ---------------- K_m = K_base + diag(sigma^2)  (LD 1024 -> LDK) -----------
__global__ void formk_kernel(const float* __restrict__ Kb,
                             const float* __restrict__ s2_all,
                             float* __restrict__ Kms, int base) {
    const int m   = blockIdx.x >> 10;       // grid = M*1024 blocks
    const int row = blockIdx.x & 1023;
    const int c4  = threadIdx.x;            // 256 threads * float4 = one row
    const float4* src = (const float4*)(Kb + (size_t)row * NPT);
    float4 v = src[c4];
    if ((row >> 2) == c4) {
        const float s2 = s2_all[(base + m) * NPT + row];
        ((float*)&v)[row & 3] += s2;
    }
    float4* dst = (float4*)(Kms + (size_t)m * MATSZ + (size_t)row * LDK);
    dst[c4] = v;
}

// ---------------- blocked Cholesky, one workgroup per matrix ---------------
// 256 threads = 8 wave32. NB=16. Trailing update on v_wmma_f32_16x16x4_f32.
__global__ void __launch_bounds__(256) chol_kernel(float* __restrict__ Kms) {
    extern __shared__ float panel[];              // 64*256 floats = 64 KB
    __shared__ float Sd[16 * 17];                 // diag tile, padded
    float* Km = Kms + (size_t)blockIdx.x * MATSZ;

    const int tid  = threadIdx.x;
    const int wave = __builtin_amdgcn_readfirstlane(tid >> 5);  // SGPR-uniform
    const int lane = tid & 31;
    const int half = lane >> 4;
    const int l16  = lane & 15;

    for (int k = 0; k < NBLK; ++k) {
        const int k16 = k * 16;
        // ---- phase 1: factor 16x16 diagonal tile in LDS ----
        {
            const int r = tid >> 4, c = tid & 15;
            Sd[r * 17 + c] = Km[(size_t)(k16 + r) * LDK + k16 + c];
        }
        __syncthreads();
        for (int c = 0; c < 16; ++c) {
            if (tid == 0) Sd[c * 17 + c] = sqrtf(Sd[c * 17 + c]);
            __syncthreads();
            if (tid > c && tid < 16) Sd[tid * 17 + c] /= Sd[c * 17 + c];
            __syncthreads();
            {
                const int r = tid & 15, j = tid >> 4;
                if (j > c && r >= j) Sd[r * 17 + j] -= Sd[r * 17 + c] * Sd[j * 17 + c];
            }
            __syncthreads();
        }
        {
            const int r = tid >> 4, c = tid & 15;
            Km[(size_t)(k16 + r) * LDK + k16 + c] = (r >= c) ? Sd[r * 17 + c] : 0.0f;
        }
        __syncthreads();

        const int nb = NBLK - 1 - k;   // remaining row blocks
        // ---- phase 2: TRSM  X * Lkk^T = A[i,k]  -> panel (LDS) + global ----
        for (int bb = wave * 2 + half; bb < nb; bb += 16) {
            const int i = k + 1 + bb;
            const size_t rowbase = (size_t)(i * 16 + l16) * LDK + k16;
            float x[16];
#pragma unroll
            for (int c = 0; c < 16; ++c) {
                float v = Km[rowbase + c];
#pragma unroll
                for (int t = 0; t < 16; ++t)
                    if (t < c) v -= x[t] * Sd[c * 17 + t];
                x[c] = v / Sd[c * 17 + c];
            }
#pragma unroll
            for (int c = 0; c < 16; ++c) {
                panel[bb * 256 + l16 * 16 + c] = x[c];
                Km[rowbase + c] = x[c];
            }
        }
        __syncthreads();

        // ---- phase 3: trailing SYRK  A[i,j] -= Xi * Xj^T  via WMMA ----
        // Guard-free 2x2 macro-tiles: phantom sub-tiles (block >= nb, or the
        // upper tile of diagonal macros) write only padding / never-read
        // upper triangle. Straight-line body, EXEC all-ones throughout.
        const int mb = (nb + 1) >> 1;
        const int mcount = mb * (mb + 1) / 2;
        for (int t = wave; t < mcount; t += 8) {
            int A = 0;
            while (((A + 1) * (A + 2)) / 2 <= t) ++A;  // triangular decode
            const int B = t - (A * (A + 1)) / 2;
            const int a0 = 2 * A, b0 = 2 * B;          // panel block indices
            const int i0 = k + 1 + a0, j0 = k + 1 + b0;

            v8f c00, c01, c10, c11;
            // C/D layout: m = v + 8*half, n = lane%16
#pragma unroll
            for (int v = 0; v < 8; ++v) {
                const size_t r0 = (size_t)(i0 * 16 + v + 8 * half) * LDK;
                const size_t r1 = (size_t)((i0 + 1) * 16 + v + 8 * half) * LDK;
                c00[v] = Km[r0 + j0 * 16 + l16];
                c01[v] = Km[r0 + (j0 + 1) * 16 + l16];
                c10[v] = Km[r1 + j0 * 16 + l16];
                c11[v] = Km[r1 + (j0 + 1) * 16 + l16];
            }

#pragma unroll
            for (int kc = 0; kc < 4; ++kc) {
                const int kcol = kc * 4 + 2 * half;     // A/B frag: k = 2*half + v
                v2f a0f, a1f, b0f, b1f;
                a0f.x = -panel[a0 * 256 + l16 * 16 + kcol];     // pre-negate A:
                a0f.y = -panel[a0 * 256 + l16 * 16 + kcol + 1]; // f32 WMMA has CNeg only
                a1f.x = -panel[(a0 + 1) * 256 + l16 * 16 + kcol];
                a1f.y = -panel[(a0 + 1) * 256 + l16 * 16 + kcol + 1];
                b0f.x =  panel[b0 * 256 + l16 * 16 + kcol];
                b0f.y =  panel[b0 * 256 + l16 * 16 + kcol + 1];
                b1f.x =  panel[(b0 + 1) * 256 + l16 * 16 + kcol];
                b1f.y =  panel[(b0 + 1) * 256 + l16 * 16 + kcol + 1];

                c00 = __builtin_amdgcn_wmma_f32_16x16x4_f32(
                        false, a0f, false, b0f, (short)0, c00, false, false);
                c01 = __builtin_amdgcn_wmma_f32_16x16x4_f32(
                        false, a0f, false, b1f, (short)0, c01, false, false);
                c10 = __builtin_amdgcn_wmma_f32_16x16x4_f32(
                        false, a1f, false, b0f, (short)0, c10, false, false);
                c11 = __builtin_amdgcn_wmma_f32_16x16x4_f32(
                        false, a1f, false, b1f, (short)0, c11, false, false);
            }

#pragma unroll
            for (int v = 0; v < 8; ++v) {
                const size_t r0 = (size_t)(i0 * 16 + v + 8 * half) * LDK;
                const size_t r1 = (size_t)((i0 + 1) * 16 + v + 8 * half) * LDK;
                Km[r0 + j0 * 16 + l16]       = c00[v];
                Km[r0 + (j0 + 1) * 16 + l16] = c01[v];
                Km[r1 + j0 * 16 + l16]       = c10[v];
                Km[r1 + (j0 + 1) * 16 + l16] = c11[v];
            }
        }
        // CU-mode: whole workgroup shares one CU/L0 -> workgroup fence is
        // sufficient for global C-tile visibility and much cheaper than agent.
        __builtin_amdgcn_fence(__ATOMIC_ACQ_REL, "workgroup");
        __syncthreads();
    }
}

// ---------------- forward solve: quad = ||L^-1 r||^2, logdet ---------------
__global__ void solve_kernel(const float* __restrict__ Kms,
                             const float* __restrict__ r_all,
                             float* __restrict__ nll, int base) {
    __shared__ float rv[NPT];
    __shared__ float Ld[16 * 17];
    __shared__ float zs[16];
    __shared__ float quad, logdet;
    const int m  = blockIdx.x;
    const int mg = base + m;
    const float* Km = Kms + (size_t)m * MATSZ;
    const int tid = threadIdx.x;

    for (int n = tid; n < NPT; n += 256) rv[n] = r_all[mg * NPT + n];
    if (tid == 0) { quad = 0.f; logdet = 0.f; }
    __syncthreads();

    for (int kb = 0; kb < NBLK; ++kb) {
        const int k16 = kb * 16;
        { const int r = tid >> 4, c = tid & 15;
          Ld[r * 17 + c] = Km[(size_t)(k16 + r) * LDK + k16 + c]; }
        __syncthreads();
        if (tid == 0) {
            float q = 0.f, ld = 0.f;
            for (int c = 0; c < 16; ++c) {
                const float z = rv[k16 + c] / Ld[c * 17 + c];
                zs[c] = z; q += z * z; ld += logf(Ld[c * 17 + c]);
                for (int r2 = c + 1; r2 < 16; ++r2)
                    rv[k16 + r2] -= Ld[r2 * 17 + c] * z;
            }
            quad += q; logdet += 2.f * ld;
        }
        __syncthreads();
        for (int n = k16 + 16 + tid; n < NPT; n += 256) {
            const float4* rowp = (const float4*)(Km + (size_t)n * LDK + k16);
            float a = 0.f;
#pragma unroll
            for (int t4 = 0; t4 < 4; ++t4) {           // 4 x B128 loads
                const float4 w = rowp[t4];
                a += w.x * zs[t4 * 4 + 0] + w.y * zs[t4 * 4 + 1]
                   + w.z * zs[t4 * 4 + 2] + w.w * zs[t4 * 4 + 3];
            }
            rv[n] -= a;
        }
        __syncthreads();
    }
    if (tid == 0)
        nll[mg] = 0.5f * (quad + logdet + (float)NPT * LOG2PI);
}

// ---------------- mean over 128 rows ---------------------------------------
__global__ void reduce_kernel(const float* __restrict__ nll, float* __restrict__ out) {
    __shared__ float s[NROWS];
    const int tid = threadIdx.x;
    s[tid] = nll[tid];
    __syncthreads();
    for (int st = 64; st > 0; st >>= 1) {
        if (tid < st) s[tid] += s[tid + st];
        __syncthreads();
    }
    if (tid == 0) out[0] = s[0] * (1.0f / (float)NROWS);
}

// ---------------------------------------------------------------------------
extern "C" void kernel_launch(void* const* d_in, const int* in_sizes, int n_in,
                              void* d_out, int out_size, void* d_ws, size_t ws_size,
                              hipStream_t stream) {
    (void)in_sizes; (void)n_in; (void)out_size;
    const float* target = (const float*)d_in[0];
    const float* mu     = (const float*)d_in[1];
    const float* sigma  = (const float*)d_in[2];
    const float* L      = (const float*)d_in[3];
    float* out = (float*)d_out;

    char* ws = (char*)d_ws;
    size_t off = 0;
    float* Kb     = (float*)(ws + off); off += (size_t)NPT * NPT * 4;      // 4 MB
    float* r_all  = (float*)(ws + off); off += (size_t)NROWS * NPT * 4;    // 512 KB
    float* s2_all = (float*)(ws + off); off += (size_t)NROWS * NPT * 4;    // 512 KB
    float* nll    = (float*)(ws + off); off += 4096;
    float* Kms    = (float*)(ws + off);

    const size_t matBytes = MATSZ * sizeof(float);       // padded slot ~4.3 MB
    const size_t avail = (ws_size > off) ? (ws_size - off) : 0;
    int CH = (int)(avail / matBytes);
    if (CH < 1) CH = 1;                 // workspace-adaptive chunking
    if (CH > NROWS) CH = NROWS;

    prep_kernel<<<NROWS, 256, 0, stream>>>(target, mu, sigma, r_all, s2_all);
    kbase_kernel<<<(NPT * NPT) / 256, 256, 0, stream>>>(L, Kb);

    const size_t cholShmem = (size_t)64 * 256 * sizeof(float);   // 64 KB panel
    for (int base = 0; base < NROWS; base += CH) {
        const int M = (NROWS - base < CH) ? (NROWS - base) : CH;
        formk_kernel<<<M * NPT, 256, 0, stream>>>(Kb, s2_all, Kms, base);
        chol_kernel<<<M, 256, cholShmem, stream>>>(Kms);
        solve_kernel<<<M, 256, 0, stream>>>(Kms, r_all, nll, base);
    }
    reduce_kernel<<<1, NROWS, 0, stream>>>(nll, out);
}